// KalmanFilterLayer_19859928776762
// MI455X (gfx1250) — compile-verified
//
#include <hip/hip_runtime.h>

// Kalman filter on gfx1250: one wave32 per (b,n) sequence, serial scan over T.
// All 16x16x16 f32 matmuls via chained V_WMMA_F32_16X16X4_F32.
// Layout identities used:
//   A-operand:  elem(M=lane%16, K=4k+2*hi+v)
//   B-operand:  elem(K=4k+2*hi+v, N=lane%16)
//   C/D:        elem(M=v+8*hi,    N=lane%16)
//   A-layout(M) == B-layout(M^T)  -> one register image of A doubles as B(A^T),
//                                    one image of Hpad doubles as B(Ht_pad).

typedef float v2f __attribute__((ext_vector_type(2)));
typedef float v8f __attribute__((ext_vector_type(8)));

#define MIN_COV 1e-6f

struct Op16 { v2f s[4]; };   // 4 K-slices of a 16x16 operand (8 VGPRs)

__device__ __forceinline__ v8f wmma16(const Op16& a, const Op16& b, v8f c) {
#pragma unroll
  for (int k = 0; k < 4; ++k)
    c = __builtin_amdgcn_wmma_f32_16x16x4_f32(false, a.s[k], false, b.s[k],
                                              (short)0, c, false, false);
  return c;
}

// load 16x16 row-major LDS tile M into WMMA-A layout
__device__ __forceinline__ Op16 ldsA(const float* M, int row, int hi) {
  Op16 o;
#pragma unroll
  for (int k = 0; k < 4; ++k) {
    int c0 = 4 * k + 2 * hi;
    v2f t; t.x = M[row * 16 + c0]; t.y = M[row * 16 + c0 + 1];
    o.s[k] = t;
  }
  return o;
}

// load 16x16 row-major LDS tile M into WMMA-B layout
__device__ __forceinline__ Op16 ldsB(const float* M, int col, int hi) {
  Op16 o;
#pragma unroll
  for (int k = 0; k < 4; ++k) {
    int r0 = 4 * k + 2 * hi;
    v2f t; t.x = M[r0 * 16 + col]; t.y = M[(r0 + 1) * 16 + col];
    o.s[k] = t;
  }
  return o;
}

// store C-layout accumulator to row-major 16x16 LDS tile
__device__ __forceinline__ void stC(float* M, v8f c, int col, int hi) {
#pragma unroll
  for (int v = 0; v < 8; ++v) M[(v + 8 * hi) * 16 + col] = c[v];
}

__global__ __launch_bounds__(32) void kalman_kernel(
    const float* __restrict__ obs,  // (B,N,T,8)
    const float* __restrict__ ctl,  // (B,N,T,1)
    const float* __restrict__ Ag,   // (16,16)
    const float* __restrict__ Bpg,  // (16,1)
    const float* __restrict__ Hg,   // (8,16)
    const float* __restrict__ Qg,   // (16,16)
    const float* __restrict__ Rg,   // (8,8)
    float* __restrict__ out,        // (B,N,T,8)
    int T) {
  __shared__ float Pm[256];   // P, row-major
  __shared__ float Sc[256];   // scratch (T1 / Kn / KnH)
  __shared__ float PH[256];   // P*Ht, row-major (cols 8..15 are zero)
  __shared__ float NSI[256];  // -inv(S) padded to 16x16
  __shared__ float Ab[256];   // A for scalar matvec
  __shared__ float Hb[128];   // H (8x16) for scalar
  __shared__ float Bpb[16];
  __shared__ float xb[16], yb[8], zb[8], fb[8];
  __shared__ float Sa[128];   // augmented [S | I] 8x16

  const int lane = threadIdx.x;
  const int hi = lane >> 4;     // lane half
  const int lc = lane & 15;     // row (A-layout) or col (B/C layout)
  const long seq = blockIdx.x;  // b*N + n

  // ---- constant WMMA operands in registers ----
  Op16 Acm;   // A-layout(A)  == B-layout(A^T)
  Op16 Hcm;   // A-layout(Hpad) == B-layout(Ht_pad)
  Op16 Hbop;  // B-layout(Hpad)
#pragma unroll
  for (int k = 0; k < 4; ++k) {
#pragma unroll
    for (int v = 0; v < 2; ++v) {
      int K = 4 * k + 2 * hi + v;
      Acm.s[k][v]  = Ag[lc * 16 + K];
      Hcm.s[k][v]  = (lc < 8) ? Hg[lc * 16 + K] : 0.0f;
      Hbop.s[k][v] = (K < 8) ? Hg[K * 16 + lc] : 0.0f;
    }
  }
  v8f QcC, RpC;
#pragma unroll
  for (int v = 0; v < 8; ++v) {
    int r = v + 8 * hi;
    QcC[v] = fmaxf(Qg[r * 16 + lc], MIN_COV);
    RpC[v] = (r < 8 && lc < 8) ? fmaxf(Rg[r * 8 + lc], MIN_COV) : 0.0f;
  }

  // ---- init state: P = I, x = 0; stage A,H,Bp in LDS ----
#pragma unroll
  for (int p = 0; p < 8; ++p) {
    int e = p * 32 + lane;
    Pm[e] = ((e >> 4) == (e & 15)) ? 1.0f : 0.0f;
    Ab[e] = Ag[e];
  }
#pragma unroll
  for (int p = 0; p < 4; ++p) { int e = p * 32 + lane; Hb[e] = Hg[e]; }
  if (lane < 16) { Bpb[lane] = Bpg[lane]; xb[lane] = 0.0f; }
  __syncthreads();

  const float* obsSeq = obs + seq * (long)T * 8;
  const float* ctlSeq = ctl + seq * (long)T;
  float* outSeq = out + seq * (long)T * 8;

#pragma unroll 1
  for (int t = 0; t < T; ++t) {
    // ---------- predict x = A x + Bp u ----------
    float u = ctlSeq[t];
    float xn = 0.0f;
    if (lane < 16) {
      xn = Bpb[lane] * u;
#pragma unroll
      for (int j = 0; j < 16; ++j) xn += Ab[lane * 16 + j] * xb[j];
    }
    __syncthreads();
    if (lane < 16) xb[lane] = xn;
    __syncthreads();
    // ---------- innovation y = obs - H x ----------
    if (lane < 8) {
      float yv = obsSeq[t * 8 + lane];
#pragma unroll
      for (int j = 0; j < 16; ++j) yv -= Hb[lane * 16 + j] * xb[j];
      yb[lane] = yv;
    }

    // ---------- predict P = A P A^T + Qc ----------
    v8f zc = {};
    Op16 Pb = ldsB(Pm, lc, hi);
    v8f t1 = wmma16(Acm, Pb, zc);        // T1 = A*P
    __syncthreads();
    stC(Sc, t1, lc, hi);
    __syncthreads();
    Op16 T1a = ldsA(Sc, lc, hi);
    v8f Pp = wmma16(T1a, Acm, QcC);      // Ppred = T1*A^T + Qc
    stC(Pm, Pp, lc, hi);
    __syncthreads();
    // ---------- PHt = P * Ht_pad ----------
    Op16 Pa = ldsA(Pm, lc, hi);
    v8f pht = wmma16(Pa, Hcm, zc);
    stC(PH, pht, lc, hi);
    __syncthreads();
    // ---------- S = Hpad * PHt + Rpad ----------
    Op16 PHb = ldsB(PH, lc, hi);
    v8f s = wmma16(Hcm, PHb, RpC);
    __syncthreads();
    if (hi == 0 && lc < 8) {             // stash [S | I] augmented
#pragma unroll
      for (int v = 0; v < 8; ++v) {
        Sa[v * 16 + lc] = s[v];
        Sa[v * 16 + 8 + lc] = (v == lc) ? 1.0f : 0.0f;
      }
    }
    __syncthreads();

    // ---------- Gauss-Jordan: Sa -> [I | S^-1] ----------
#pragma unroll 1
    for (int k = 0; k < 8; ++k) {
      float ip = 1.0f / Sa[k * 16 + k];
      float rk = 0.0f, fr = 0.0f;
      if (lane < 16) rk = Sa[k * 16 + lane] * ip;
      if (lane < 8) fr = Sa[lane * 16 + k];
      __syncthreads();
      if (lane < 16) Sa[k * 16 + lane] = rk;
      if (lane < 8) fb[lane] = fr;
      __syncthreads();
#pragma unroll
      for (int p = 0; p < 4; ++p) {
        int e = p * 32 + lane, r = e >> 4, c = e & 15;
        if (r != k) Sa[r * 16 + c] -= fb[r] * Sa[k * 16 + c];
      }
      __syncthreads();
    }

    // ---------- z = S^-1 y ; NSI = -S^-1 padded ----------
    if (lane < 8) {
      float z = 0.0f;
#pragma unroll
      for (int j = 0; j < 8; ++j) z += Sa[lane * 16 + 8 + j] * yb[j];
      zb[lane] = z;
    }
#pragma unroll
    for (int p = 0; p < 8; ++p) {
      int e = p * 32 + lane, r = e >> 4, c = e & 15;
      NSI[e] = (r < 8 && c < 8) ? -Sa[r * 16 + 8 + c] : 0.0f;
    }
    __syncthreads();
    // ---------- x += PHt z ; emit x[:8] ----------
    float xacc = 0.0f;
    if (lane < 16) {
      xacc = xb[lane];
#pragma unroll
      for (int j = 0; j < 8; ++j) xacc += PH[lane * 16 + j] * zb[j];
    }
    __syncthreads();
    if (lane < 16) xb[lane] = xacc;
    if (lane < 8) outSeq[t * 8 + lane] = xacc;
    __syncthreads();

    // ---------- P = Ppred - K H Ppred  (K = PHt S^-1) ----------
    Op16 PHa = ldsA(PH, lc, hi);
    Op16 NSb = ldsB(NSI, lc, hi);
    v8f kn = wmma16(PHa, NSb, zc);       // Kn = PHt * (-S^-1) = -K
    __syncthreads();
    stC(Sc, kn, lc, hi);
    __syncthreads();
    Op16 Kna = ldsA(Sc, lc, hi);
    v8f knh = wmma16(Kna, Hbop, zc);     // KnH = -K * Hpad
    __syncthreads();
    stC(Sc, knh, lc, hi);
    __syncthreads();
    Op16 KHa = ldsA(Sc, lc, hi);
    Op16 Ppb = ldsB(Pm, lc, hi);
    v8f Pn = wmma16(KHa, Ppb, Pp);       // Pnew = (-KH)*Ppred + Ppred
    __syncthreads();
    stC(Pm, Pn, lc, hi);
    __syncthreads();
  }
}

extern "C" void kernel_launch(void* const* d_in, const int* in_sizes, int n_in,
                              void* d_out, int out_size, void* d_ws, size_t ws_size,
                              hipStream_t stream) {
  const float* obs = (const float*)d_in[0];  // (8,128,512,8)
  const float* ctl = (const float*)d_in[1];  // (8,128,512,1)
  const float* A   = (const float*)d_in[2];  // (16,16)
  const float* Bp  = (const float*)d_in[3];  // (16,1)
  const float* H   = (const float*)d_in[4];  // (8,16)
  const float* Q   = (const float*)d_in[5];  // (16,16)
  const float* R   = (const float*)d_in[6];  // (8,8)
  float* out = (float*)d_out;

  const int T = 512;
  const int BN = in_sizes[1] / T;  // control has B*N*T elements -> 1024 sequences
  kalman_kernel<<<BN, 32, 0, stream>>>(obs, ctl, A, Bp, H, Q, R, out, T);
}